// SentenceGather_90288802497333
// MI455X (gfx1250) — compile-verified
//
#include <hip/hip_runtime.h>
#include <hip/hip_bf16.h>
#include <stdint.h>

// Problem constants (from the reference)
#define BB   16
#define LL   8192
#define DD   768
#define SS   64

#define TPB     192                 // DD / 4 floats -> one float4 lane slice per thread
#define CH      8                   // rows per async chunk
#define ROWB    (DD * 4)            // 3072 bytes per row
#define CHUNKB  (CH * ROWB)         // 24576 bytes per chunk
#define LDSB    (2 * CHUNKB)        // 49152 bytes, double buffered

__global__ __launch_bounds__(TPB)
void seg_mean_kernel(const float* __restrict__ x,
                     const int*   __restrict__ sidx,
                     float*       __restrict__ out)
{
    const int s   = blockIdx.x;   // segment id   [0, 64)
    const int b   = blockIdx.y;   // batch sample [0, 16)
    const int tid = threadIdx.x;

    // ---- binary search the contiguous run [start, end) of segment s ----
    const int* sb = sidx + (size_t)b * LL;
    int lo = 0, hi = LL;
    while (lo < hi) { int mid = (lo + hi) >> 1; if (sb[mid] <  s) lo = mid + 1; else hi = mid; }
    const int start = lo;
    hi = LL;
    while (lo < hi) { int mid = (lo + hi) >> 1; if (sb[mid] <= s) lo = mid + 1; else hi = mid; }
    const int end   = lo;
    const int count = end - start;

    const float*   xb       = x + (size_t)b * LL * DD;
    const uint32_t lane_off = (uint32_t)tid * 16u;
    const uint32_t lds_base = __builtin_amdgcn_groupstaticsize(); // dynamic-LDS start

    extern __shared__ float4 lds[];   // LDSB bytes, same region the asm writes

    float4 acc = make_float4(0.f, 0.f, 0.f, 0.f);

    const int nfull = count / CH;

    // Issue one chunk (8 rows) of per-lane b128 async copies global -> LDS.
    auto issue_chunk = [&](int c, int buf) {
        uint32_t g = (uint32_t)(start + c * CH) * (uint32_t)ROWB + lane_off; // < 25 MB, fits i32
        uint32_t l = lds_base + (uint32_t)buf * (uint32_t)CHUNKB + lane_off;
        #pragma unroll
        for (int r = 0; r < CH; ++r) {
            asm volatile("global_load_async_to_lds_b128 %0, %1, %2"
                         :: "v"(l), "v"(g), "s"(xb)
                         : "memory");
            g += ROWB;
            l += ROWB;
        }
    };

    if (nfull > 0) issue_chunk(0, 0);
    if (nfull > 1) issue_chunk(1, 1);

    for (int c = 0; c < nfull; ++c) {
        if (c + 1 < nfull) {
            // chunk c+1 (8 async ops) may remain in flight; chunk c is done (in-order)
            asm volatile("s_wait_asynccnt 8" ::: "memory");
        } else {
            asm volatile("s_wait_asynccnt 0" ::: "memory");
        }
        const int buf = c & 1;
        #pragma unroll
        for (int r = 0; r < CH; ++r) {
            float4 v = lds[(size_t)(buf * CH + r) * TPB + tid];
            acc.x += v.x; acc.y += v.y; acc.z += v.z; acc.w += v.w;
        }
        if (c + 2 < nfull) {
            // WAR guard: make sure our DS reads of this buffer retired before
            // the async engine starts overwriting it.
            asm volatile("s_wait_dscnt 0" ::: "memory");
            issue_chunk(c + 2, buf);
        }
    }

    // ---- tail rows (count % CH) via direct float4 loads ----
    for (int r = start + nfull * CH; r < end; ++r) {
        const float4 v = *((const float4*)(xb + (size_t)r * DD) + tid);
        acc.x += v.x; acc.y += v.y; acc.z += v.z; acc.w += v.w;
    }

    const float inv = 1.0f / fmaxf((float)count, 1.0f);
    float4 res = make_float4(acc.x * inv, acc.y * inv, acc.z * inv, acc.w * inv);
    *((float4*)(out + ((size_t)b * SS + s) * DD) + tid) = res;
}

extern "C" void kernel_launch(void* const* d_in, const int* in_sizes, int n_in,
                              void* d_out, int out_size, void* d_ws, size_t ws_size,
                              hipStream_t stream) {
    (void)in_sizes; (void)n_in; (void)out_size; (void)d_ws; (void)ws_size;
    const float* x    = (const float*)d_in[0];
    const int*   sidx = (const int*)d_in[1];
    // d_in[2] = num_segments scalar (fixed to 64 here)
    float* out = (float*)d_out;

    dim3 grid(SS, BB);
    seg_mean_kernel<<<grid, TPB, LDSB, stream>>>(x, sidx, out);
}